// AgnosticNMS_807453851765
// MI455X (gfx1250) — compile-verified
//
#include <hip/hip_runtime.h>
#include <hip/hip_bf16.h>
#include <stdint.h>

// ---------------------------------------------------------------------------
// CDNA5 (gfx1250) helpers: async global->LDS copies + ASYNCcnt wait
// ---------------------------------------------------------------------------
#define AS_G __attribute__((address_space(1)))
#define AS_L __attribute__((address_space(3)))

typedef int v4i __attribute__((ext_vector_type(4)));

#define NEG_INF (-__builtin_inff())

__device__ __forceinline__ void async_g2l_b128(const void* gsrc, void* ldst) {
#if defined(__HIP_DEVICE_COMPILE__)
#if __has_builtin(__builtin_amdgcn_global_load_async_to_lds_b128)
  const AS_G void* g = (const AS_G void*)gsrc;  // generic -> AS(1)
  AS_L void*       l = (AS_L void*)ldst;        // generic -> AS(3)
  __builtin_amdgcn_global_load_async_to_lds_b128(
      (AS_G v4i*)g, (AS_L v4i*)l, 0, 0);
#else
  unsigned lds_off = (unsigned)(uintptr_t)((AS_L void*)ldst);
  unsigned long long gaddr = (unsigned long long)(uintptr_t)gsrc;
  asm volatile("global_load_async_to_lds_b128 %0, %1, off"
               :: "v"(lds_off), "v"(gaddr) : "memory");
#endif
#endif
}

__device__ __forceinline__ void wait_async_zero() {
#if defined(__HIP_DEVICE_COMPILE__)
#if __has_builtin(__builtin_amdgcn_s_wait_asynccnt)
  __builtin_amdgcn_s_wait_asynccnt(0);
#else
  asm volatile("s_wait_asynccnt 0" ::: "memory");
#endif
#endif
}

namespace {
constexpr int   kN        = 25200;   // boxes per batch
constexpr int   kC        = 80;      // classes
constexpr int   kTopK     = 100;
constexpr float kConf     = 0.25f;
constexpr float kIouT     = 0.45f;
constexpr int   kP1Rows   = 256;     // rows (boxes) per phase-1 block
constexpr int   kP1Stride = 84;      // padded LDS row stride in dwords (16B aligned)
constexpr int   kP2T      = 1024;    // phase-2 threads (32 wave32 waves)
}

// ---------------------------------------------------------------------------
// Phase 1: sc[b,n] = max_c scores[b,n,c].
// Stage a 256x80 f32 tile into LDS with async b128 copies, reduce from LDS.
// (classes argmax is done LAZILY in phase 2 for the <=800 selected rows only,
// cutting total DRAM traffic roughly in half.)
// ---------------------------------------------------------------------------
__global__ __launch_bounds__(kP1Rows) void nms_p1_scoremax(
    const float* __restrict__ scores, float* __restrict__ sc_out, int total_rows) {
  __shared__ __align__(16) float tile[kP1Rows * kP1Stride];

  const int tid  = threadIdx.x;
  const int row0 = blockIdx.x * kP1Rows;
  constexpr int kChunksPerRow = kC / 4;  // 20 float4 chunks per row

  // Async-stage the tile: global -> LDS without VGPR round trips.
  for (int chunk = tid; chunk < kP1Rows * kChunksPerRow; chunk += kP1Rows) {
    const int r   = chunk / kChunksPerRow;
    const int c4  = chunk % kChunksPerRow;
    const int row = row0 + r;
    if (row < total_rows) {
      async_g2l_b128(scores + (size_t)row * kC + c4 * 4,
                     tile + r * kP1Stride + c4 * 4);
    }
  }
  wait_async_zero();   // each wave drains its own ASYNCcnt...
  __syncthreads();     // ...then all waves rendezvous -> whole tile visible

  const int row = row0 + tid;
  if (row < total_rows) {
    const float* p = tile + tid * kP1Stride;
    float m = p[0];
#pragma unroll
    for (int c = 1; c < kC; ++c) m = fmaxf(m, p[c]);
    sc_out[row] = m;
  }
}

// ---------------------------------------------------------------------------
// Phase 2: one workgroup per batch runs the sequential greedy NMS.
// live[] scores in LDS; argmax = per-thread scan + wave32 shuffle reduce +
// cross-wave LDS reduce; boxes stay L2-resident (403KB/batch, L2 = 192MB).
// ---------------------------------------------------------------------------
__global__ __launch_bounds__(kP2T) void nms_p2_greedy(
    const float* __restrict__ boxes,     // [B,N,4] y1,x1,y2,x2
    const float* __restrict__ classes,   // [B,N,C]
    const float* __restrict__ sc_ws,     // [B,N] from phase 1
    float* __restrict__ out, int B) {
  __shared__ __align__(16) float live[kN];
  __shared__ float red_val[kP2T / 32];
  __shared__ int   red_idx[kP2T / 32];
  __shared__ float sel_box[4];
  __shared__ int   sel_idx_s;
  __shared__ int   sel_valid_s;
  __shared__ int   selk_idx[kTopK];
  __shared__ float selk_val[kTopK];
  __shared__ int   selk_valid[kTopK];

  const int b    = blockIdx.x;
  const int tid  = threadIdx.x;
  const int lane = tid & 31;
  const int wave = tid >> 5;
  const float* sc_b    = sc_ws + (size_t)b * kN;
  const float* boxes_b = boxes + (size_t)b * kN * 4;

  // Async-stage this batch's scores into LDS (25200 floats = 6300 b128 copies).
  for (int q = tid; q < kN / 4; q += kP2T)
    async_g2l_b128(sc_b + q * 4, live + q * 4);
  wait_async_zero();
  __syncthreads();

  // Confidence threshold in place.
  for (int i = tid; i < kN; i += kP2T) {
    const float v = live[i];
    live[i] = (v >= kConf) ? v : NEG_INF;
  }
  __syncthreads();

  for (int k = 0; k < kTopK; ++k) {
    // ---- global argmax over live[] (first-max tie-break = lowest index) ----
    float bv = NEG_INF;
    int   bi = 0x7FFFFFFF;
    for (int i = tid; i < kN; i += kP2T) {
      const float v = live[i];
      if (v > bv || (v == bv && i < bi)) { bv = v; bi = i; }
    }
#pragma unroll
    for (int off = 16; off > 0; off >>= 1) {
      const float ov = __shfl_xor(bv, off, 32);
      const int   oi = __shfl_xor(bi, off, 32);
      if (ov > bv || (ov == bv && oi < bi)) { bv = ov; bi = oi; }
    }
    if (lane == 0) { red_val[wave] = bv; red_idx[wave] = bi; }
    __syncthreads();
    if (wave == 0) {
      constexpr int nw = kP2T / 32;
      float v = (lane < nw) ? red_val[lane] : NEG_INF;
      int   i = (lane < nw) ? red_idx[lane] : 0x7FFFFFFF;
#pragma unroll
      for (int off = 16; off > 0; off >>= 1) {
        const float ov = __shfl_xor(v, off, 32);
        const int   oi = __shfl_xor(i, off, 32);
        if (ov > v || (ov == v && oi < i)) { v = ov; i = oi; }
      }
      if (lane == 0) {
        const int valid = (v > NEG_INF) ? 1 : 0;
        sel_idx_s   = i;
        sel_valid_s = valid;
        selk_idx[k] = i;
        selk_val[k] = v;
        selk_valid[k] = valid;
        const float4 bx = *reinterpret_cast<const float4*>(boxes_b + (size_t)i * 4);
        sel_box[0] = bx.x; sel_box[1] = bx.y; sel_box[2] = bx.z; sel_box[3] = bx.w;
      }
    }
    __syncthreads();

    // ---- suppress: IoU(selected, all) > thr, plus the selected box itself ----
    if (sel_valid_s) {
      const float sy1 = sel_box[0], sx1 = sel_box[1];
      const float sy2 = sel_box[2], sx2 = sel_box[3];
      const float area1 = (sy2 - sy1) * (sx2 - sx1);
      const int   si    = sel_idx_s;
      for (int i = tid; i < kN; i += kP2T) {
        if (live[i] == NEG_INF) continue;  // already dead: suppression is a no-op
        const float4 bx = *reinterpret_cast<const float4*>(boxes_b + (size_t)i * 4);
        const float ty = fmaxf(sy1, bx.x), tx = fmaxf(sx1, bx.y);
        const float by = fminf(sy2, bx.z), bxr = fminf(sx2, bx.w);
        const float ih = fmaxf(by - ty, 0.0f), iw = fmaxf(bxr - tx, 0.0f);
        const float inter = ih * iw;
        const float area2 = (bx.z - bx.x) * (bx.w - bx.y);
        const float iou = inter / (area1 + area2 - inter + 1e-9f);
        if (iou > kIouT || i == si) live[i] = NEG_INF;
      }
    }
    __syncthreads();
  }

  // ---- outputs; lazy classes-argmax only for the selected rows ----
  float* out_boxes   = out;                                   // [B,100,4]
  float* out_scores  = out + (size_t)B * kTopK * 4;           // [B,100]
  float* out_classes = out_scores + (size_t)B * kTopK;        // [B,100]
  int*   out_valid   = (int*)(out_classes + (size_t)B * kTopK);  // [B] i32

  if (tid < kTopK) {
    const int k     = tid;
    const int valid = selk_valid[k];
    const int si    = selk_idx[k];

    float4 bx;
    float  scv  = -1.0f;
    float  clsf = -1.0f;
    if (valid) {
      bx  = *reinterpret_cast<const float4*>(boxes_b + (size_t)si * 4);
      scv = selk_val[k];
      const float4* crow =
          reinterpret_cast<const float4*>(classes + ((size_t)b * kN + si) * kC);
      float m = NEG_INF;
      int   mi = 0;
#pragma unroll
      for (int c4 = 0; c4 < kC / 4; ++c4) {
        const float4 v = crow[c4];
        if (v.x > m) { m = v.x; mi = c4 * 4 + 0; }
        if (v.y > m) { m = v.y; mi = c4 * 4 + 1; }
        if (v.z > m) { m = v.z; mi = c4 * 4 + 2; }
        if (v.w > m) { m = v.w; mi = c4 * 4 + 3; }
      }
      clsf = (float)mi;
    } else {
      bx = make_float4(0.0f, 0.0f, 0.0f, 0.0f);
    }
    reinterpret_cast<float4*>(out_boxes)[(size_t)b * kTopK + k] = bx;
    out_scores[(size_t)b * kTopK + k]  = scv;
    out_classes[(size_t)b * kTopK + k] = clsf;
  }
  if (tid == 0) {
    int cnt = 0;
#pragma unroll
    for (int k = 0; k < kTopK; ++k) cnt += selk_valid[k];
    out_valid[b] = cnt;
  }
}

// ---------------------------------------------------------------------------
// Harness entry point
// ---------------------------------------------------------------------------
extern "C" void kernel_launch(void* const* d_in, const int* in_sizes, int n_in,
                              void* d_out, int out_size, void* d_ws, size_t ws_size,
                              hipStream_t stream) {
  (void)n_in; (void)out_size; (void)ws_size;
  const float* boxes   = (const float*)d_in[0];  // [B,N,4]
  const float* classes = (const float*)d_in[1];  // [B,N,C]
  const float* scores  = (const float*)d_in[2];  // [B,N,C]
  // d_in[3] = topk_all (=100), fixed by the reference; kTopK hardcoded.

  const int B          = in_sizes[0] / (kN * 4);
  const int total_rows = B * kN;
  float* sc_ws = (float*)d_ws;  // B*N floats (~800 KB)

  const int grid1 = (total_rows + kP1Rows - 1) / kP1Rows;
  nms_p1_scoremax<<<grid1, kP1Rows, 0, stream>>>(scores, sc_ws, total_rows);
  nms_p2_greedy<<<B, kP2T, 0, stream>>>(boxes, classes, sc_ws, (float*)d_out, B);
}